// EMAPrototypes_37907381354731
// MI455X (gfx1250) — compile-verified
//
#include <hip/hip_runtime.h>

typedef float v2f __attribute__((ext_vector_type(2)));
typedef float v8f __attribute__((ext_vector_type(8)));

#define NUM_C   1000
#define PAD_C   1024
#define DIMS    128
#define SLICE_W 32
#define NSLICES 4     // 4 * 32 = 128 columns
#define CHUNKS  64    // 64 chunks * 4 slices = 256 blocks
#define MOM     0.9f

// ---------------------------------------------------------------------------
// Kernel A: zero the workspace accumulators (sums [1000x128] + counts [1000]).
// d_ws is not re-poisoned between replays, so we must clear it each call.
// ---------------------------------------------------------------------------
__global__ void ema_zero_kernel(float* __restrict__ sums, unsigned* __restrict__ gcnt) {
  int i = blockIdx.x * blockDim.x + threadIdx.x;
  if (i < NUM_C * DIMS) sums[i] = 0.0f;
  if (i < NUM_C)        gcnt[i] = 0u;
}

// ---------------------------------------------------------------------------
// Kernel B: segment-sum via LDS-privatized accumulation.
// Block = (chunk, slice). LDS holds a full 1024-class x 32-col f32 accumulator
// (128 KB -> needs CDNA5's 320 KB/WGP LDS; 2 blocks per WGP).
// Each 8-lane group streams one row's 32-float slice as a float4 per lane.
// ---------------------------------------------------------------------------
__global__ void ema_accum_kernel(const int*   __restrict__ cls,
                                 const float* __restrict__ feats,
                                 float*       __restrict__ sums,
                                 unsigned*    __restrict__ gcnt,
                                 int N) {
  extern __shared__ char smem[];
  float*    acc  = (float*)smem;                         // PAD_C * SLICE_W
  unsigned* ccnt = (unsigned*)(acc + PAD_C * SLICE_W);   // PAD_C (slice 0 only)

  const int tid   = threadIdx.x;
  const int slice = blockIdx.y;

  for (int i = tid; i < PAD_C * SLICE_W; i += blockDim.x) acc[i] = 0.0f;
  if (slice == 0)
    for (int i = tid; i < PAD_C; i += blockDim.x) ccnt[i] = 0u;
  __syncthreads();

  const int g   = tid >> 3;  // 32 row-groups per 256-thread block
  const int sub = tid & 7;   // 8 lanes per row, one float4 each
  const float4* f4 = (const float4*)feats;

  for (int rb = blockIdx.x * 32; rb < N; rb += gridDim.x * 32) {
    int row = rb + g;
    if (row < N) {
      int c = cls[row];
      c = min(max(c, 0), NUM_C - 1);
      float4 v = f4[row * (DIMS / 4) + slice * (SLICE_W / 4) + sub];
      float* a = &acc[c * SLICE_W + sub * 4];
      atomicAdd(a + 0, v.x);   // ds_add_f32, bank-conflict-light layout
      atomicAdd(a + 1, v.y);
      atomicAdd(a + 2, v.z);
      atomicAdd(a + 3, v.w);
      if (slice == 0 && sub == 0) atomicAdd(&ccnt[c], 1u);
    }
  }
  __syncthreads();

  // Flush block-private partials to global (global_atomic_add_f32).
  for (int i = tid; i < NUM_C * SLICE_W; i += blockDim.x) {
    int c = i / SLICE_W, j = i % SLICE_W;
    float vsum = acc[c * SLICE_W + j];
    if (vsum != 0.0f) atomicAdd(&sums[c * DIMS + slice * SLICE_W + j], vsum);
  }
  if (slice == 0) {
    for (int i = tid; i < NUM_C; i += blockDim.x) {
      unsigned cc = ccnt[i];
      if (cc) atomicAdd(&gcnt[i], cc);
    }
  }
}

// ---------------------------------------------------------------------------
// Kernel C: EMA finalize as two diagonal GEMMs on the matrix core:
//   new_vec = Diag(a) * vec + Diag(b) * sums   (f32 WMMA 16x16x4, K chained)
// per-class scalars: absent -> a=1,b=0 ; fresh -> a=0,b=1/n ; else a=.9,b=.1/n
// One wave per 16x16 output tile; 8 waves/block cover all 8 dim-tiles.
// ---------------------------------------------------------------------------
__global__ void ema_finalize_kernel(const float*    __restrict__ vec,
                                    const int*      __restrict__ cnt,
                                    const float*    __restrict__ sums,
                                    const unsigned* __restrict__ gcnt,
                                    float*          __restrict__ out) {
  const int tid  = threadIdx.x;
  const int dt   = tid >> 5;   // wave id = dim tile 0..7
  const int lane = tid & 31;
  const int half = lane >> 4;
  const int m    = lane & 15;  // A row index AND B/D column index for this lane
  const int ct   = blockIdx.x; // class tile 0..62 (class 992..1007 padded)

  const int cm = min(ct * 16 + m, NUM_C - 1);
  const float countf  = (float)gcnt[cm];
  const bool  present = countf > 0.0f;
  const float inv     = 1.0f / fmaxf(countf, 1.0f);
  const bool  fresh   = (cnt[cm] == 0);
  const float a_m = present ? (fresh ? 0.0f : MOM) : 1.0f;
  const float b_m = present ? (fresh ? inv : (1.0f - MOM) * inv) : 0.0f;

  const int col = dt * 16 + m;

  v8f c = {};
  // D += Diag(a) * vec   (K = 0..15 in four K=4 slices)
  #pragma unroll
  for (int k = 0; k < 4; ++k) {
    const int kk0 = 4 * k + 2 * half;   // A layout: lanes16-31 hold K+2,K+3
    const int kk1 = kk0 + 1;
    v2f A, B;
    A.x = (m == kk0) ? a_m : 0.0f;
    A.y = (m == kk1) ? a_m : 0.0f;
    const int r0 = min(ct * 16 + kk0, NUM_C - 1);
    const int r1 = min(ct * 16 + kk1, NUM_C - 1);
    B.x = vec[r0 * DIMS + col];
    B.y = vec[r1 * DIMS + col];
    c = __builtin_amdgcn_wmma_f32_16x16x4_f32(false, A, false, B,
                                              (short)0, c, false, false);
  }
  // D += Diag(b) * sums
  #pragma unroll
  for (int k = 0; k < 4; ++k) {
    const int kk0 = 4 * k + 2 * half;
    const int kk1 = kk0 + 1;
    v2f A, B;
    A.x = (m == kk0) ? b_m : 0.0f;
    A.y = (m == kk1) ? b_m : 0.0f;
    const int r0 = min(ct * 16 + kk0, NUM_C - 1);
    const int r1 = min(ct * 16 + kk1, NUM_C - 1);
    B.x = sums[r0 * DIMS + col];
    B.y = sums[r1 * DIMS + col];
    c = __builtin_amdgcn_wmma_f32_16x16x4_f32(false, A, false, B,
                                              (short)0, c, false, false);
  }

  // D layout: VGPR v, lane half -> class row v + 8*half, column = lane&15
  #pragma unroll
  for (int v = 0; v < 8; ++v) {
    const int crow = ct * 16 + v + 8 * half;
    if (crow < NUM_C) out[crow * DIMS + col] = c[v];
  }

  // new_cnt = cnt + counts, stored as f32 (exact: counts < 2^24)
  if (dt == 0 && half == 0) {
    const int crow = ct * 16 + m;
    if (crow < NUM_C)
      out[NUM_C * DIMS + crow] = (float)(cnt[crow] + (int)gcnt[crow]);
  }
}

// ---------------------------------------------------------------------------
extern "C" void kernel_launch(void* const* d_in, const int* in_sizes, int n_in,
                              void* d_out, int out_size, void* d_ws, size_t ws_size,
                              hipStream_t stream) {
  const int*   cls   = (const int*)d_in[0];
  const float* feats = (const float*)d_in[1];
  const float* vec   = (const float*)d_in[2];
  const int*   cnt   = (const int*)d_in[3];
  const int N = in_sizes[0];

  float*    sums = (float*)d_ws;                                   // 1000*128 f32
  unsigned* gcnt = (unsigned*)((char*)d_ws + NUM_C * DIMS * 4);    // 1000 u32
  float*    out  = (float*)d_out;

  ema_zero_kernel<<<(NUM_C * DIMS + 255) / 256, 256, 0, stream>>>(sums, gcnt);

  const size_t smem = (size_t)(PAD_C * SLICE_W + PAD_C) * 4;  // 135168 B (< 320 KB/WGP)
  dim3 gridB(CHUNKS, NSLICES);
  ema_accum_kernel<<<gridB, 256, smem, stream>>>(cls, feats, sums, gcnt, N);

  ema_finalize_kernel<<<(NUM_C + 15) / 16, 256, 0, stream>>>(vec, cnt, sums, gcnt, out);
}